// AttentionHook_36361193128593
// MI455X (gfx1250) — compile-verified
//
#include <hip/hip_runtime.h>
#include <hip/hip_bf16.h>

typedef __attribute__((ext_vector_type(2))) float v2f;
typedef __attribute__((ext_vector_type(8))) float v8f;

#define LOG2E 1.4426950408889634f

constexpr int Bb = 8;
constexpr int Cc = 256;
constexpr int Nn = 4096;   // 64*64

// ---------------------------------------------------------------------------
// Phase 1: f[b,n] = sum_c wq[c]*x[b,c,n]; g2[b,n] = log2e * sum_c wk[c]*x
// ---------------------------------------------------------------------------
__global__ void fg_kernel(const float* __restrict__ x, const float* __restrict__ wq,
                          const float* __restrict__ wk, float* __restrict__ f,
                          float* __restrict__ g2) {
    __shared__ float swq[256], swk[256];
    const int tid = threadIdx.x;
    swq[tid] = wq[tid];
    swk[tid] = wk[tid];
    __syncthreads();
    const int blocksPerB = Nn / 256;                 // 16
    const int b = blockIdx.x / blocksPerB;
    const int n = (blockIdx.x % blocksPerB) * 256 + tid;
    const float* xb = x + (size_t)b * Cc * Nn + n;
    float accf = 0.0f, accg = 0.0f;
#pragma unroll 8
    for (int c = 0; c < Cc; ++c) {
        float xv = xb[(size_t)c * Nn];
        accf = fmaf(swq[c], xv, accf);
        accg = fmaf(swk[c], xv, accg);
    }
    f[b * Nn + n]  = accf;
    g2[b * Nn + n] = accg * LOG2E;
}

// ---------------------------------------------------------------------------
// Phase 1b: per-batch max/min of f (needed for numerically safe softmax)
// ---------------------------------------------------------------------------
__global__ void minmax_kernel(const float* __restrict__ f, float* __restrict__ fmm) {
    __shared__ float smax[256], smin[256];
    const int b = blockIdx.x, tid = threadIdx.x;
    float mx = -3.4e38f, mn = 3.4e38f;
    for (int i = tid; i < Nn; i += 256) {
        float v = f[b * Nn + i];
        mx = fmaxf(mx, v);
        mn = fminf(mn, v);
    }
    smax[tid] = mx; smin[tid] = mn;
    __syncthreads();
    for (int s = 128; s > 0; s >>= 1) {
        if (tid < s) {
            smax[tid] = fmaxf(smax[tid], smax[tid + s]);
            smin[tid] = fminf(smin[tid], smin[tid + s]);
        }
        __syncthreads();
    }
    if (tid == 0) { fmm[2 * b] = smax[0]; fmm[2 * b + 1] = smin[0]; }
}

// ---------------------------------------------------------------------------
// Phase 2: per column m: c2[m] = max_n f[n]*g2[m]; rz[m] = (1-gamma)/Z[m]
// ---------------------------------------------------------------------------
__global__ void z_kernel(const float* __restrict__ f, const float* __restrict__ g2,
                         const float* __restrict__ fmm, const float* __restrict__ gamma,
                         float* __restrict__ c2, float* __restrict__ rz) {
    __shared__ float sf[256];
    const int blocksPerB = Nn / 256;
    const int b = blockIdx.x / blocksPerB;
    const int m = (blockIdx.x % blocksPerB) * 256 + threadIdx.x;
    const float gm = g2[b * Nn + m];
    const float cm = (gm >= 0.0f) ? gm * fmm[2 * b] : gm * fmm[2 * b + 1];
    float z = 0.0f;
    for (int nc = 0; nc < Nn; nc += 256) {
        __syncthreads();
        sf[threadIdx.x] = f[b * Nn + nc + threadIdx.x];
        __syncthreads();
#pragma unroll 8
        for (int i = 0; i < 256; ++i)
            z += exp2f(fmaf(sf[i], gm, -cm));      // terms <= 1, Z <= 4096
    }
    c2[b * Nn + m] = cm;
    rz[b * Nn + m] = (1.0f - gamma[0]) / z;
}

// ---------------------------------------------------------------------------
// Phase 3: h[b] = wv(256x256) * x[b](256xN) via V_WMMA_F32_16X16X4_F32
// Workgroup = 512 threads (16 waves), strip = [256 rows x 64 cols]
// ---------------------------------------------------------------------------
__global__ __launch_bounds__(512) void h_kernel(const float* __restrict__ x,
                                                const float* __restrict__ wv,
                                                float* __restrict__ h) {
    __shared__ __align__(16) float lA[256 * 34];   // wv rows, stride-34 pad
    __shared__ __align__(16) float lB[64 * 34];    // x^T tile: [n][k]
    const int tid  = threadIdx.x;
    const int wave = tid >> 5, lane = tid & 31, l16 = lane & 15, hi = lane >> 4;
    const int strips = Nn / 64;                    // 64
    const int b  = blockIdx.x / strips;
    const int m0 = (blockIdx.x % strips) * 64;
    const float* xb = x + (size_t)b * Cc * Nn;

    v8f acc[4] = {};
    for (int kc = 0; kc < Cc; kc += 32) {
#pragma unroll
        for (int i = tid; i < 256 * 32; i += 512) {  // stage wv chunk
            int r = i >> 5, c = i & 31;
            lA[r * 34 + c] = wv[r * Cc + kc + c];
        }
#pragma unroll
        for (int i = tid; i < 32 * 64; i += 512) {   // stage x^T chunk
            int k = i >> 6, n = i & 63;
            lB[n * 34 + k] = xb[(size_t)(kc + k) * Nn + m0 + n];
        }
        __syncthreads();
#pragma unroll
        for (int k = 0; k < 32; k += 4) {
            // A 16x4 layout: lanes 0-15 -> K=0,1 ; lanes 16-31 -> K=2,3
            v2f va = *(const v2f*)&lA[(16 * wave + l16) * 34 + k + 2 * hi];
#pragma unroll
            for (int t = 0; t < 4; ++t) {
                v2f vb = *(const v2f*)&lB[(16 * t + l16) * 34 + k + 2 * hi];
                acc[t] = __builtin_amdgcn_wmma_f32_16x16x4_f32(
                    false, va, false, vb, (short)0, acc[t], false, false);
            }
        }
        __syncthreads();
    }
    float* hb = h + (size_t)b * Cc * Nn;
#pragma unroll
    for (int t = 0; t < 4; ++t) {
        int col = m0 + 16 * t + l16;
#pragma unroll
        for (int r = 0; r < 8; ++r) {
            int row = 16 * wave + r + 8 * hi;      // C/D layout: M = r + 8*hi
            hb[(size_t)row * Nn + col] = acc[t][r];
        }
    }
}

// ---------------------------------------------------------------------------
// Phase 4: out[b,c,m] = rz[m] * sum_n h[c,n]*exp2(f[n]*g2[m]-c2[m]) + gam*x
// Beta tile is computed ONCE per workgroup per K-chunk into LDS (16x fewer
// transcendentals than per-wave recompute), then consumed by WMMA.
// ---------------------------------------------------------------------------
__global__ __launch_bounds__(512) void attn_kernel(const float* __restrict__ x,
        const float* __restrict__ h,  const float* __restrict__ f,
        const float* __restrict__ g2, const float* __restrict__ c2,
        const float* __restrict__ rz, const float* __restrict__ gamma,
        float* __restrict__ out) {
    __shared__ __align__(16) float lA[256 * 34];   // h chunk  [256 c][32 n]
    __shared__ __align__(16) float lB[64 * 34];    // beta^T   [64 m][32 n]
    const int tid  = threadIdx.x;
    const int wave = tid >> 5, lane = tid & 31, l16 = lane & 15, hi = lane >> 4;
    const int strips = Nn / 64;
    const int b  = blockIdx.x / strips;
    const int m0 = (blockIdx.x % strips) * 64;

    // beta staging assignment: thread fills lB[mbase+16j][nloc], j=0..3
    const int nloc  = tid & 31;
    const int mbase = tid >> 5;                    // 0..15
    float gm[4], cm[4];
#pragma unroll
    for (int j = 0; j < 4; ++j) {
        int m = m0 + mbase + 16 * j;
        gm[j] = g2[b * Nn + m];
        cm[j] = c2[b * Nn + m];
    }

    v8f acc[4] = {};
    const float* hb = h + (size_t)b * Cc * Nn;
    const float* fb = f + b * Nn;

    for (int nc = 0; nc < Nn; nc += 32) {
#pragma unroll
        for (int i = tid; i < 256 * 32; i += 512) {  // stage h chunk
            int r = i >> 5, c = i & 31;
            lA[r * 34 + c] = hb[(size_t)r * Nn + nc + c];
        }
        if (nc + 32 < Nn)                            // global_prefetch_b8
            __builtin_prefetch(&hb[(size_t)(tid >> 5) * Nn + nc + 32 + (tid & 31)], 0, 0);
        // stage beta chunk: beta[m][n] = exp2(f[n]*g2[m] - c2[m])
        float fv = fb[nc + nloc];
#pragma unroll
        for (int j = 0; j < 4; ++j)
            lB[(mbase + 16 * j) * 34 + nloc] = exp2f(fmaf(fv, gm[j], -cm[j]));
        __syncthreads();
#pragma unroll
        for (int k = 0; k < 32; k += 4) {
            v2f va = *(const v2f*)&lA[(16 * wave + l16) * 34 + k + 2 * hi];
#pragma unroll
            for (int t = 0; t < 4; ++t) {
                v2f vb = *(const v2f*)&lB[(16 * t + l16) * 34 + k + 2 * hi];
                acc[t] = __builtin_amdgcn_wmma_f32_16x16x4_f32(
                    false, va, false, vb, (short)0, acc[t], false, false);
            }
        }
        __syncthreads();
    }

    const float gam = gamma[0];
    const float* xb = x + (size_t)b * Cc * Nn;
    float* ob = out + (size_t)b * Cc * Nn;
#pragma unroll
    for (int t = 0; t < 4; ++t) {
        int col = m0 + 16 * t + l16;
        float rzv = rz[b * Nn + col];
#pragma unroll
        for (int r = 0; r < 8; ++r) {
            int row = 16 * wave + r + 8 * hi;
            size_t idx = (size_t)row * Nn + col;
            ob[idx] = fmaf(acc[t][r], rzv, gam * xb[idx]);
        }
    }
}

// ---------------------------------------------------------------------------
extern "C" void kernel_launch(void* const* d_in, const int* in_sizes, int n_in,
                              void* d_out, int out_size, void* d_ws, size_t ws_size,
                              hipStream_t stream) {
    const float* x     = (const float*)d_in[0];
    const float* wq    = (const float*)d_in[1];
    const float* wk    = (const float*)d_in[2];
    const float* wv    = (const float*)d_in[3];
    const float* gamma = (const float*)d_in[4];
    float* out = (float*)d_out;

    // workspace layout (floats): h [B*C*N] | f | g2 | c2 | rz  [B*N each] | fmm [2B]
    float* ws  = (float*)d_ws;
    float* h   = ws;
    float* f   = h  + (size_t)Bb * Cc * Nn;
    float* g2  = f  + (size_t)Bb * Nn;
    float* c2v = g2 + (size_t)Bb * Nn;
    float* rzv = c2v + (size_t)Bb * Nn;
    float* fmm = rzv + (size_t)Bb * Nn;

    fg_kernel    <<<Bb * (Nn / 256), 256, 0, stream>>>(x, wq, wk, f, g2);
    minmax_kernel<<<Bb,              256, 0, stream>>>(f, fmm);
    z_kernel     <<<Bb * (Nn / 256), 256, 0, stream>>>(f, g2, fmm, gamma, c2v, rzv);
    h_kernel     <<<Bb * (Nn / 64),  512, 0, stream>>>(x, wv, h);
    attn_kernel  <<<Bb * (Nn / 64),  512, 0, stream>>>(x, h, f, g2, c2v, rzv, gamma, out);
}